// ExpMinProcessor_72859825210033
// MI455X (gfx1250) — compile-verified
//
#include <hip/hip_runtime.h>
#include <math.h>

#define BATCH_N 256
#define VOCAB_N 128000
#define TPB 512
#define BOOST_V 50.0f

#if defined(__AMDGCN__) && __has_builtin(__builtin_amdgcn_global_load_async_to_lds_b128) && __has_builtin(__builtin_amdgcn_s_wait_asynccnt)
#define USE_ASYNC 1
typedef __attribute__((__vector_size__(16))) int v4i_t;
typedef __attribute__((address_space(1))) v4i_t* g128_t;   // global b128 pointer
typedef __attribute__((address_space(3))) v4i_t* l128_t;   // LDS b128 pointer
#endif

// Monotonic float -> uint key (larger logit => larger key)
__device__ __forceinline__ unsigned key_of(float x) {
  unsigned u = __float_as_uint(x);
  return (u & 0x80000000u) ? ~u : (u | 0x80000000u);
}

// Deterministic integer quantization of exp(l - m); e in (0,1] -> q in [0, 2^32]
__device__ __forceinline__ unsigned long long quant_e(float l, float m) {
  float e = __expf(l - m);
  return (unsigned long long)(e * 4294967296.0f);
}

extern "C" __global__ __launch_bounds__(TPB)
void expmin_topp_kernel(const float* __restrict__ logits,
                        const float* __restrict__ xi,
                        float* __restrict__ out,
                        int* __restrict__ winners) {
  const int row = blockIdx.x;
  const int tid = threadIdx.x;
  const float* L = logits + (size_t)row * VOCAB_N;
  const float* X = xi + (size_t)row * VOCAB_N;
  float* O = out + (size_t)row * VOCAB_N;
  const float4* L4 = reinterpret_cast<const float4*>(L);
  const float4* X4 = reinterpret_cast<const float4*>(X);
  float4* O4 = reinterpret_cast<float4*>(O);

  __shared__ unsigned long long s_hist[2048];   // 16 KB, weighted radix histogram
  __shared__ unsigned long long s_part[TPB];    //  4 KB, per-thread scan partials
  __shared__ float s_redf[TPB];
  __shared__ int s_redi[TPB];
  __shared__ unsigned long long s_budget;
  __shared__ int s_sel;
#ifdef USE_ASYNC
  __shared__ float s_lb[2][TPB * 4];            // 16 KB double buffer (logits)
  __shared__ float s_xb[2][TPB * 4];            // 16 KB double buffer (xi)
#endif

  // ---- Phase A: row max (for softmax stabilization) ----
  float m = -INFINITY;
  for (int i = tid; i < VOCAB_N / 4; i += TPB) {
    float4 v = L4[i];
    m = fmaxf(m, fmaxf(fmaxf(v.x, v.y), fmaxf(v.z, v.w)));
  }
  s_redf[tid] = m;
  __syncthreads();
  for (int off = TPB / 2; off > 0; off >>= 1) {
    if (tid < off) s_redf[tid] = fmaxf(s_redf[tid], s_redf[tid + off]);
    __syncthreads();
  }
  const float mrow = s_redf[0];
  __syncthreads();

  // ---- Phase B: weighted radix-select for boundary key c ----
  // keep(v) <=> CumLE(key_v) > (1 - TOP_P) * Z, exact in integer-quantized weights.
  unsigned prefix = 0u, prefMask = 0u;
  for (int st = 0; st < 3; ++st) {
    const int shift = (st == 0) ? 21 : (st == 1) ? 10 : 0;
    const int nb = (st == 2) ? 1024 : 2048;
    const int bpt = nb / TPB;                  // bins per thread: 4,4,2
    for (int j = tid; j < nb; j += TPB) s_hist[j] = 0ull;
    __syncthreads();
    for (int i = tid; i < VOCAB_N / 4; i += TPB) {
      float4 v = L4[i];
#define HGR(LV) { float l_ = (LV); unsigned k_ = key_of(l_);                       \
      if ((k_ & prefMask) == prefix)                                               \
        atomicAdd(&s_hist[(k_ >> shift) & (unsigned)(nb - 1)], quant_e(l_, mrow)); }
      HGR(v.x) HGR(v.y) HGR(v.z) HGR(v.w)
#undef HGR
    }
    __syncthreads();

    // previous stage's remaining budget (valid for st>0); read before overwrite
    const unsigned long long U_prev = s_budget;

    // per-thread partial over its contiguous bins (cache bin values in regs)
    unsigned long long hreg[4];
    unsigned long long partial = 0ull;
    const int myb = tid * bpt;
    for (int k = 0; k < bpt; ++k) { hreg[k] = s_hist[myb + k]; partial += hreg[k]; }
    s_part[tid] = partial;
    __syncthreads();
    // Hillis-Steele inclusive scan over s_part (u64)
    for (int off = 1; off < TPB; off <<= 1) {
      unsigned long long v = s_part[tid];
      unsigned long long add = (tid >= off) ? s_part[tid - off] : 0ull;
      __syncthreads();
      s_part[tid] = v + add;
      __syncthreads();
    }
    const unsigned long long incl = s_part[tid];
    const unsigned long long base = incl - partial;        // exclusive prefix
    const unsigned long long total = s_part[TPB - 1];
    const unsigned long long U =
        (st == 0) ? (unsigned long long)((double)total * 0.1) : U_prev;

    if (tid == 0) s_sel = -1;
    __syncthreads();   // also orders all U_prev reads before s_budget overwrite

    // unique crossing bin: base' <= U < base' + h  (exact integer arithmetic)
    unsigned long long acc = base;
    for (int k = 0; k < bpt; ++k) {
      unsigned long long h = hreg[k];
      if (acc <= U && acc + h > U) { s_sel = myb + k; s_budget = U - acc; }
      acc += h;
    }
    __syncthreads();
    if (s_sel < 0) {   // unreachable numerical guard (uniform branch)
      if (tid == 0) {
        int found = 0; unsigned long long nbud = 0ull;
        for (int j = nb - 1; j >= 0; --j)
          if (s_hist[j] != 0ull) { found = j; nbud = s_hist[j] - 1ull; break; }
        s_sel = found; s_budget = nbud;
      }
      __syncthreads();
    }
    prefix |= ((unsigned)s_sel) << shift;
    prefMask |= ((unsigned)(nb - 1)) << shift;
  }
  const unsigned ckey = prefix;  // exact 32-bit boundary key; keep iff key >= ckey

  // ---- Phase C: fused copy + exp-min argmin over the nucleus ----
  float best = INFINITY;
  int bestIdx = 0x7fffffff;
  // score = log(-log(xi)) - logit  (monotone transform of -log(xi)/p)
#define PROC(LV, XV, II) { float l_ = (LV); unsigned k_ = key_of(l_);              \
  if (k_ >= ckey) { float sc_ = __logf(0.0f - __logf(XV)) - l_;                    \
    if (sc_ < best || (sc_ == best && (II) < bestIdx)) { best = sc_; bestIdx = (II); } } }

#ifdef USE_ASYNC
  const int TILE = TPB * 4;           // 2048 floats per tile
  const int NT = VOCAB_N / TILE;      // 62 full tiles, 1024-element tail
  {
    const float* gl = L + tid * 4;
    const float* gx = X + tid * 4;
    __builtin_amdgcn_global_load_async_to_lds_b128(
        (g128_t)gl, (l128_t)&s_lb[0][tid * 4], 0, 0);
    __builtin_amdgcn_global_load_async_to_lds_b128(
        (g128_t)gx, (l128_t)&s_xb[0][tid * 4], 0, 0);
  }
  for (int t = 0; t < NT; ++t) {
    const int cur = t & 1;
    const int nxt = cur ^ 1;
    if (t + 1 < NT) {
      const float* gl = L + (t + 1) * TILE + tid * 4;
      const float* gx = X + (t + 1) * TILE + tid * 4;
      __builtin_amdgcn_global_load_async_to_lds_b128(
          (g128_t)gl, (l128_t)&s_lb[nxt][tid * 4], 0, 0);
      __builtin_amdgcn_global_load_async_to_lds_b128(
          (g128_t)gx, (l128_t)&s_xb[nxt][tid * 4], 0, 0);
      __builtin_amdgcn_s_wait_asynccnt(2);   // oldest 2 (this wave's tile t) done
    } else {
      __builtin_amdgcn_s_wait_asynccnt(0);
    }
    float4 lv = *reinterpret_cast<float4*>(&s_lb[cur][tid * 4]);
    float4 xv = *reinterpret_cast<float4*>(&s_xb[cur][tid * 4]);
    const int gbase = t * TILE + tid * 4;
    O4[gbase >> 2] = lv;
    PROC(lv.x, xv.x, gbase + 0)
    PROC(lv.y, xv.y, gbase + 1)
    PROC(lv.z, xv.z, gbase + 2)
    PROC(lv.w, xv.w, gbase + 3)
  }
  for (int i = NT * TILE + tid; i < VOCAB_N; i += TPB) {   // tail
    float l = L[i]; float x = X[i];
    O[i] = l;
    PROC(l, x, i)
  }
#else
  for (int i = tid; i < VOCAB_N / 4; i += TPB) {
    float4 lv = L4[i];
    float4 xv = X4[i];
    O4[i] = lv;
    const int gbase = i * 4;
    PROC(lv.x, xv.x, gbase + 0)
    PROC(lv.y, xv.y, gbase + 1)
    PROC(lv.z, xv.z, gbase + 2)
    PROC(lv.w, xv.w, gbase + 3)
  }
#endif
#undef PROC

  // ---- block argmin reduction (score, index) ----
  s_redf[tid] = best;
  s_redi[tid] = bestIdx;
  __syncthreads();
  for (int off = TPB / 2; off > 0; off >>= 1) {
    if (tid < off) {
      float a = s_redf[tid], b = s_redf[tid + off];
      int ia = s_redi[tid], ib = s_redi[tid + off];
      if (b < a || (b == a && ib < ia)) { s_redf[tid] = b; s_redi[tid] = ib; }
    }
    __syncthreads();
  }
  if (tid == 0) winners[row] = s_redi[0];
}

extern "C" __global__ void apply_boost_kernel(const float* __restrict__ logits,
                                              float* __restrict__ out,
                                              const int* __restrict__ winners) {
  int r = blockIdx.x * blockDim.x + threadIdx.x;
  if (r < BATCH_N) {
    int idx = winners[r];
    if (idx >= 0 && idx < VOCAB_N) {
      size_t off = (size_t)r * VOCAB_N + (size_t)idx;
      out[off] = logits[off] + BOOST_V;
    }
  }
}

extern "C" void kernel_launch(void* const* d_in, const int* in_sizes, int n_in,
                              void* d_out, int out_size, void* d_ws, size_t ws_size,
                              hipStream_t stream) {
  (void)in_sizes; (void)n_in; (void)out_size; (void)ws_size;
  // d_in[0] = input_ids (int64, unused), d_in[1] = logits (f32), d_in[2] = xi (f32)
  const float* logits = (const float*)d_in[1];
  const float* xi = (const float*)d_in[2];
  float* out = (float*)d_out;
  int* winners = (int*)d_ws;   // 256 ints of scratch

  expmin_topp_kernel<<<dim3(BATCH_N), dim3(TPB), 0, stream>>>(logits, xi, out, winners);
  apply_boost_kernel<<<dim3(1), dim3(BATCH_N), 0, stream>>>(logits, out, winners);
}